// SBLEST_70884140253746
// MI455X (gfx1250) — compile-verified
//
#include <hip/hip_runtime.h>
#include <hip/hip_bf16.h>

// ---------------------------------------------------------------------------
// SBLEST on MI455X (gfx1250).  All GEMMs go through the CDNA5 matrix pipe
// via V_WMMA_F32_16X16X4_F32 (full f32, matches reference precision).
// Single unified GEMM kernel: 32x32 tile per wave (2x2 accumulators),
// branch-free clamped loads (OOB rows/cols only feed OOB outputs, which the
// store guard drops), b64 vector loads on contiguous fragments.
// Everything is L2-resident (<200MB), so this is matrix-pipe bound.
// ---------------------------------------------------------------------------

typedef float v2f __attribute__((ext_vector_type(2)));
typedef float v8f __attribute__((ext_vector_type(8)));

// Problem constants (match reference)
#define NC   32      // channels
#define NT   750     // time samples
#define NM   200     // trials
#define NK   5       // delays
#define KC   160     // NK*NC
#define TP   768     // padded time (multiple of 32)
#define KC2  25600   // KC*KC
#define NIT  15
#define JAC_SWEEPS 10

// scratch scalar slots (floats)
#define S_LAM    0
#define S_LOSSO  1
#define S_DONE   2
#define S_COMMIT 3
#define S_THETA  4
#define S_RES2   5
#define S_LOGDET 6
#define S_LDOT   7
#define S_MAXABS 8

// ---------------------------------------------------------------------------
// Batched WMMA GEMM: C[b] = opA(A[b]) * opB(B[b]).   Requires K%4==0.
// opA: TA? A^T : A  (A stored M x K row-major, lda; TA => A stored K x M).
// One wave computes a 32x32 block with 4 accumulators; A/B fragments are
// each reused twice.  Load indices are clamped to [0, Md/Nd) instead of
// zero-guarded: an out-of-range A row only contributes to an out-of-range
// C row (same for B cols), and those stores are masked.  This keeps every
// load unconditional -> s_clause'd b64 loads, no exec-mask churn, and the
// 4 independent WMMAs cover the WMMA->WMMA RAW hazard window.
// ---------------------------------------------------------------------------
template<bool TA, bool TB>
__global__ __launch_bounds__(32)
void gemm_wmma_t(const float* __restrict__ Ab, long sA, int lda,
                 const float* __restrict__ Bb, long sB, int ldb,
                 float* __restrict__ Cb, long sC, int ldc,
                 int Md, int Nd, int Kd)
{
  const float* A = Ab + (long)blockIdx.z * sA;
  const float* B = Bb + (long)blockIdx.z * sB;
  float*       Cp = Cb + (long)blockIdx.z * sC;
  const int lane = (int)threadIdx.x;
  const int half = lane >> 4;          // 0: lanes 0-15, 1: lanes 16-31
  const int l15  = lane & 15;
  // clamped load rows/cols (tile 0 and tile 1 of the 2x2 block)
  const int r0 = min(blockIdx.y * 32 + l15,      Md - 1);
  const int r1 = min(blockIdx.y * 32 + l15 + 16, Md - 1);
  const int c0 = min(blockIdx.x * 32 + l15,      Nd - 1);
  const int c1 = min(blockIdx.x * 32 + l15 + 16, Nd - 1);
  v8f acc00 = {}, acc01 = {}, acc10 = {}, acc11 = {};

#pragma unroll 2
  for (int kk = 0; kk < Kd; kk += 4) {
    const int ka = kk + half * 2;      // lanes 0-15: K={kk,kk+1}; 16-31: {kk+2,kk+3}
    v2f a0, a1, b0, b1;
    if (TA) {
      a0.x = A[(long)(ka    )*lda + r0]; a0.y = A[(long)(ka + 1)*lda + r0];
      a1.x = A[(long)(ka    )*lda + r1]; a1.y = A[(long)(ka + 1)*lda + r1];
    } else {
      a0 = *(const v2f*)(A + (long)r0 * lda + ka);
      a1 = *(const v2f*)(A + (long)r1 * lda + ka);
    }
    if (TB) {
      b0 = *(const v2f*)(B + (long)c0 * ldb + ka);
      b1 = *(const v2f*)(B + (long)c1 * ldb + ka);
    } else {
      b0.x = B[(long)(ka    )*ldb + c0]; b0.y = B[(long)(ka + 1)*ldb + c0];
      b1.x = B[(long)(ka    )*ldb + c1]; b1.y = B[(long)(ka + 1)*ldb + c1];
    }
    acc00 = __builtin_amdgcn_wmma_f32_16x16x4_f32(false, a0, false, b0, (short)0, acc00, false, false);
    acc01 = __builtin_amdgcn_wmma_f32_16x16x4_f32(false, a0, false, b1, (short)0, acc01, false, false);
    acc10 = __builtin_amdgcn_wmma_f32_16x16x4_f32(false, a1, false, b0, (short)0, acc10, false, false);
    acc11 = __builtin_amdgcn_wmma_f32_16x16x4_f32(false, a1, false, b1, (short)0, acc11, false, false);
  }

  // C/D layout: VGPR g -> row = rowbase + g + 8*half, col = colbase + l15
  const int cs0 = blockIdx.x * 32 + l15;       // store cols (unclamped)
  const int cs1 = cs0 + 16;
#pragma unroll
  for (int g = 0; g < 8; ++g) {
    const int rs0 = blockIdx.y * 32 + g + half * 8;   // store rows (unclamped)
    const int rs1 = rs0 + 16;
    if (rs0 < Md) {
      if (cs0 < Nd) Cp[(long)rs0 * ldc + cs0] = acc00[g];
      if (cs1 < Nd) Cp[(long)rs0 * ldc + cs1] = acc01[g];
    }
    if (rs1 < Md) {
      if (cs0 < Nd) Cp[(long)rs1 * ldc + cs0] = acc10[g];
      if (cs1 < Nd) Cp[(long)rs1 * ldc + cs1] = acc11[g];
    }
  }
}

// ---------------------------------------------------------------------------
// Pack X (C,T,M) into delayed-embedded Xp (M, KC, TP), zero padded.
// Xhat[k*C+c, t, m] = X[c, t-k, m] for t>=k, t<750; else 0.
// ---------------------------------------------------------------------------
__global__ void pack_xhat(const float* __restrict__ X, float* __restrict__ Xp) {
  long idx = (long)blockIdx.x * 256 + threadIdx.x;
  const long total = (long)NM * KC * TP;
  if (idx >= total) return;
  int t = (int)(idx % TP);
  long r = idx / TP;
  int a = (int)(r % KC);
  int m = (int)(r / KC);
  int k = a / NC, c = a % NC;
  float v = 0.f;
  int ts = t - k;
  if (t < NT && ts >= 0) v = X[((long)c * NT + ts) * NM + m];
  Xp[idx] = v;
}

// per-trial trace normalize
__global__ void trace_scale(float* __restrict__ R) {
  float* Rm = R + (long)blockIdx.x * KC2;
  __shared__ float red[256];
  float p = 0.f;
  for (int a = threadIdx.x; a < KC; a += 256) p += Rm[a * KC + a];
  red[threadIdx.x] = p; __syncthreads();
  for (int s = 128; s > 0; s >>= 1) { if ((int)threadIdx.x < s) red[threadIdx.x] += red[threadIdx.x + s]; __syncthreads(); }
  float inv = 1.f / red[0];
  for (int q = threadIdx.x; q < KC2; q += 256) Rm[q] *= inv;
}

__global__ void mean_wh(const float* __restrict__ R, float* __restrict__ Wh) {
  int p = blockIdx.x * 256 + threadIdx.x;
  if (p >= KC2) return;
  float s = 0.f;
  for (int m = 0; m < NM; ++m) s += R[(long)m * KC2 + p];
  Wh[p] = s / (float)NM;
}

__global__ void copy_k(const float* __restrict__ src, float* __restrict__ dst, long n) {
  long i = (long)blockIdx.x * 256 + threadIdx.x;
  if (i < n) dst[i] = src[i];
}

// batched symmetrize: out[b] = (in[b] + in[b]^T)/2, KC x KC
__global__ void symavg(const float* __restrict__ in, float* __restrict__ out, int batch) {
  long idx = (long)blockIdx.x * 256 + threadIdx.x;
  long total = (long)batch * KC2;
  if (idx >= total) return;
  long b = idx / KC2; int p = (int)(idx % KC2);
  int a = p / KC, c = p % KC;
  const float* ib = in + b * KC2;
  out[idx] = 0.5f * (ib[a * KC + c] + ib[c * KC + a]);
}

// ---------------------------------------------------------------------------
// Batched cyclic Jacobi eigh (symmetric), one block per matrix, in global mem.
// A (in-place, destroyed -> used as temp), V = eigenvectors (columns),
// w = eigenvalues sorted ascending (matches jnp.linalg.eigh ordering).
// ---------------------------------------------------------------------------
__global__ void jacobi_eigh(float* __restrict__ Ab, float* __restrict__ Vb,
                            float* __restrict__ wb, int n, int nsweep) {
  float* A = Ab + (long)blockIdx.x * n * n;
  float* V = Vb + (long)blockIdx.x * n * n;
  float* w = wb + (long)blockIdx.x * n;
  const int tid = threadIdx.x, nt = blockDim.x;
  for (int i = tid; i < n * n; i += nt) V[i] = (i / n == i % n) ? 1.f : 0.f;
  __shared__ float sc, ss;
  __shared__ int sskip;
  __syncthreads();
  for (int sw = 0; sw < nsweep; ++sw) {
    for (int p = 0; p < n - 1; ++p) {
      for (int q = p + 1; q < n; ++q) {
        if (tid == 0) {
          float apq = A[p * n + q];
          if (fabsf(apq) < 1e-12f) { sskip = 1; }
          else {
            sskip = 0;
            float app = A[p * n + p], aqq = A[q * n + q];
            float th = (aqq - app) / (2.f * apq);
            float t = (th >= 0.f ? 1.f : -1.f) / (fabsf(th) + sqrtf(th * th + 1.f));
            float c = 1.f / sqrtf(t * t + 1.f);
            sc = c; ss = t * c;
          }
        }
        __syncthreads();
        if (!sskip) {
          float c = sc, s = ss;
          for (int i = tid; i < n; i += nt) {         // A <- A*J (columns p,q)
            float aip = A[i * n + p], aiq = A[i * n + q];
            A[i * n + p] = c * aip - s * aiq;
            A[i * n + q] = s * aip + c * aiq;
          }
          __syncthreads();
          for (int i = tid; i < n; i += nt) {         // A <- J^T*A (rows p,q); V <- V*J
            float api = A[p * n + i], aqi = A[q * n + i];
            A[p * n + i] = c * api - s * aqi;
            A[q * n + i] = s * api + c * aqi;
            float vip = V[i * n + p], viq = V[i * n + q];
            V[i * n + p] = c * vip - s * viq;
            V[i * n + q] = s * vip + c * viq;
          }
        }
        __syncthreads();
      }
    }
  }
  // sort eigenvalues ascending, permute V columns (A reused as temp)
  __shared__ int   perm[256];
  __shared__ float dv[256];
  if (tid == 0) {
    for (int i = 0; i < n; ++i) { dv[i] = A[i * n + i]; perm[i] = i; }
    for (int i = 0; i < n - 1; ++i) {
      int mi = i;
      for (int j = i + 1; j < n; ++j) if (dv[perm[j]] < dv[perm[mi]]) mi = j;
      int t2 = perm[i]; perm[i] = perm[mi]; perm[mi] = t2;
    }
  }
  __syncthreads();
  for (int i = tid; i < n; i += nt) w[i] = dv[perm[i]];
  for (int idx = tid; idx < n * n; idx += nt) {
    int r = idx / n, cl = idx % n;
    A[idx] = V[r * n + perm[cl]];
  }
  __syncthreads();
  for (int idx = tid; idx < n * n; idx += nt) V[idx] = A[idx];
}

// Whinv = Q * diag(w^-1/2) * Q^T
__global__ void whinvhalf(const float* __restrict__ Q, const float* __restrict__ w,
                          float* __restrict__ Whinv) {
  int p = blockIdx.x * 256 + threadIdx.x;
  if (p >= KC2) return;
  int a = p / KC, b = p % KC;
  float s = 0.f;
  for (int k = 0; k < KC; ++k) s += Q[a * KC + k] * rsqrtf(w[k]) * Q[b * KC + k];
  Whinv[p] = s;
}

// Qlw[m][a][k] = Q[m][a][k] * log(w[m][k])
__global__ void qlw_scale(const float* __restrict__ Q, const float* __restrict__ w,
                          float* __restrict__ Qlw) {
  long idx = (long)blockIdx.x * 256 + threadIdx.x;
  if (idx >= (long)NM * KC2) return;
  long m = idx / KC2; int r = (int)(idx % KC2); int k = r % KC;
  Qlw[idx] = Q[idx] * logf(w[m * KC + k]);
}

// Sigma[i][i] += lam
__global__ void add_lam_diag(float* __restrict__ Sg, const float* __restrict__ scal) {
  int i = threadIdx.x + blockIdx.x * 256;
  if (i < NM) Sg[i * NM + i] += scal[S_LAM];
}

// one-block Cholesky: factor (into L buf), logdet, dense inverse into Inv
__global__ void chol_invert(const float* __restrict__ Sg, float* __restrict__ L,
                            float* __restrict__ Inv, float* __restrict__ scal) {
  const int n = NM; const int tid = threadIdx.x;
  for (int i = tid; i < n * n; i += 256) L[i] = Sg[i];
  __syncthreads();
  for (int k = 0; k < n; ++k) {
    if (tid == 0) L[k * n + k] = sqrtf(L[k * n + k]);
    __syncthreads();
    float dk = L[k * n + k];
    for (int i = k + 1 + tid; i < n; i += 256) L[i * n + k] /= dk;
    __syncthreads();
    for (int j = k + 1 + tid; j < n; j += 256) {
      float ljk = L[j * n + k];
      for (int i = j; i < n; ++i) L[i * n + j] -= L[i * n + k] * ljk;
    }
    __syncthreads();
  }
  if (tid == 0) {
    float s = 0.f;
    for (int k = 0; k < n; ++k) s += logf(L[k * n + k]);
    scal[S_LOGDET] = 2.f * s;
  }
  __syncthreads();
  for (int col = tid; col < n; col += 256) {
    for (int i = 0; i < col; ++i) Inv[i * n + col] = 0.f;   // y_i = 0, i < col
    for (int i = col; i < n; ++i) {                          // forward: L y = e_col
      float s = (i == col) ? 1.f : 0.f;
      for (int j = col; j < i; ++j) s -= L[i * n + j] * Inv[j * n + col];
      Inv[i * n + col] = s / L[i * n + i];
    }
    for (int i = n - 1; i >= 0; --i) {                       // backward: L^T x = y
      float s = Inv[i * n + col];
      for (int j = i + 1; j < n; ++j) s -= L[j * n + i] * Inv[j * n + col];
      Inv[i * n + col] = s / L[i * n + i];
    }
  }
}

// y2 = SyInv @ labels ; scal[LDOT] = labels . y2
__global__ void y2_loss(const float* __restrict__ SyInv, const float* __restrict__ lab,
                        float* __restrict__ y2, float* __restrict__ scal) {
  __shared__ float red[256];
  int tid = threadIdx.x;
  float contrib = 0.f;
  if (tid < NM) {
    float s = 0.f;
    for (int k = 0; k < NM; ++k) s += SyInv[tid * NM + k] * lab[k];
    y2[tid] = s;
    contrib = lab[tid] * s;
  }
  red[tid] = contrib; __syncthreads();
  for (int s = 128; s > 0; s >>= 1) { if (tid < s) red[tid] += red[tid + s]; __syncthreads(); }
  if (tid == 0) scal[S_LDOT] = red[0];
}

// uc[p] = sum_m RtP[m][p] * y2[m]
__global__ void uc_k(const float* __restrict__ RtP, const float* __restrict__ y2,
                     float* __restrict__ uc) {
  int p = blockIdx.x * 256 + threadIdx.x;
  if (p >= KC2) return;
  float s = 0.f;
  for (int m = 0; m < NM; ++m) s += RtP[(long)m * KC2 + p] * y2[m];
  uc[p] = s;
}

__global__ void unew_k(const float* __restrict__ uc, float* __restrict__ Unew) {
  int p = blockIdx.x * 256 + threadIdx.x;
  if (p >= KC2) return;
  int j = p / KC, b = p % KC;
  Unew[p] = 0.5f * (uc[p] + uc[b * KC + j]);
}

// Phi[j] = Psi - T2[j]  (in place on T2 buffer)
__global__ void phi_fin(const float* __restrict__ Psi, float* __restrict__ PhiB) {
  long idx = (long)blockIdx.x * 256 + threadIdx.x;
  if (idx >= (long)KC * KC2) return;
  int p = (int)(idx % KC2);
  PhiB[idx] = Psi[p] - PhiB[idx];
}

__global__ void phisum_k(const float* __restrict__ PhiB, float* __restrict__ PHI) {
  int p = blockIdx.x * 256 + threadIdx.x;
  if (p >= KC2) return;
  float s = 0.f;
  for (int j = 0; j < KC; ++j) s += PhiB[(long)j * KC2 + p];
  PHI[p] = s;
}

__global__ void psinew_k(const float* __restrict__ UU, const float* __restrict__ PHI,
                         float* __restrict__ PsiNew) {
  int p = blockIdx.x * 256 + threadIdx.x;
  if (p >= KC2) return;
  int a = p / KC, b = p % KC;
  float uu = 0.5f * (UU[p] + UU[b * KC + a]);
  float ph = 0.5f * (PHI[p] + PHI[b * KC + a]);
  PsiNew[p] = (uu + ph) / (float)KC;
}

// theta += sum Phi .* Graw   (Graw symmetric so elementwise dot is exact)
__global__ void theta_k(const float* __restrict__ PhiB, const float* __restrict__ Graw,
                        float* __restrict__ scal) {
  __shared__ float red[256];
  long idx = (long)blockIdx.x * 256 + threadIdx.x;
  float v = 0.f;
  if (idx < (long)KC * KC2) v = PhiB[idx] * Graw[idx];
  red[threadIdx.x] = v; __syncthreads();
  for (int s = 128; s > 0; s >>= 1) { if ((int)threadIdx.x < s) red[threadIdx.x] += red[threadIdx.x + s]; __syncthreads(); }
  if (threadIdx.x == 0) atomicAdd(&scal[S_THETA], red[0]);
}

// resid_m = labels[m] - R_train[m,:] . vec(Unew) ; res2 += resid^2
__global__ void resid_k(const float* __restrict__ Rtr, const float* __restrict__ Unew,
                        const float* __restrict__ lab, float* __restrict__ scal) {
  __shared__ float red[256];
  int m = blockIdx.x;
  const float* row = Rtr + (long)m * KC2;
  float s = 0.f;
  for (int p = threadIdx.x; p < KC2; p += 256) s += row[p] * Unew[p];
  red[threadIdx.x] = s; __syncthreads();
  for (int st = 128; st > 0; st >>= 1) { if ((int)threadIdx.x < st) red[threadIdx.x] += red[threadIdx.x + st]; __syncthreads(); }
  if (threadIdx.x == 0) {
    float r = lab[m] - red[0];
    atomicAdd(&scal[S_RES2], r * r);
  }
}

__global__ void zero_iter(float* __restrict__ scal) {
  scal[S_THETA] = 0.f; scal[S_RES2] = 0.f;
}

__global__ void init_em(float* __restrict__ Psi, float* __restrict__ U,
                        float* __restrict__ scal) {
  int p = blockIdx.x * 256 + threadIdx.x;
  if (p < KC2) {
    Psi[p] = (p / KC == p % KC) ? 1.f : 0.f;
    U[p] = 0.f;
  }
  if (p == 0) { scal[S_LAM] = 1.f; scal[S_LOSSO] = 1e12f; scal[S_DONE] = 0.f; }
}

// end-of-iteration scan-step semantics (freeze once converged)
__global__ void finalize_iter(float* __restrict__ scal) {
  float theta = scal[S_THETA], res2 = scal[S_RES2];
  float logdet = scal[S_LOGDET], ldot = scal[S_LDOT];
  float loss = ldot + logdet;
  float lam_new = (res2 + theta) / (float)NM;
  float done_old = scal[S_DONE];
  float delta = fabsf(scal[S_LOSSO] - loss) / fabsf(scal[S_LOSSO]);
  scal[S_COMMIT] = (done_old == 0.f) ? 1.f : 0.f;   // commit uses OLD done
  if (done_old == 0.f) { scal[S_LAM] = lam_new; scal[S_LOSSO] = loss; }
  if (delta < 2e-4f) scal[S_DONE] = 1.f;            // new_done = done | (delta<TOL)
}

__global__ void commit_mats(float* __restrict__ Psi, const float* __restrict__ PsiNew,
                            float* __restrict__ U, const float* __restrict__ Unew,
                            const float* __restrict__ scal) {
  int p = blockIdx.x * 256 + threadIdx.x;
  if (p >= KC2) return;
  if (scal[S_COMMIT] > 0.5f) { Psi[p] = PsiNew[p]; U[p] = Unew[p]; }
}

__global__ void maxabs_k(const float* __restrict__ d, float* __restrict__ scal) {
  __shared__ float red[256];
  float m = 0.f;
  for (int i = threadIdx.x; i < KC; i += 256) m = fmaxf(m, fabsf(d[i]));
  red[threadIdx.x] = m; __syncthreads();
  for (int s = 128; s > 0; s >>= 1) { if ((int)threadIdx.x < s) red[threadIdx.x] = fmaxf(red[threadIdx.x], red[threadIdx.x + s]); __syncthreads(); }
  if (threadIdx.x == 0) scal[S_MAXABS] = red[0];
}

// d_out = [W(25600) | alpha(160) | V_sel(25600) | Wh(25600)]
__global__ void final_out(const float* __restrict__ U, const float* __restrict__ Wh,
                          const float* __restrict__ d, const float* __restrict__ V,
                          const float* __restrict__ scal, float* __restrict__ out) {
  int p = blockIdx.x * 256 + threadIdx.x;
  if (p >= KC2) return;
  float mx = scal[S_MAXABS];
  out[p] = U[p];
  out[KC2 + KC + KC2 + p] = Wh[p];
  int col = p % KC;
  bool mask = (fabsf(d[col]) / mx) > 0.05f;
  out[KC2 + KC + p] = mask ? V[p] : 0.f;
  if (p < KC) out[KC2 + p] = ((fabsf(d[p]) / mx) > 0.05f) ? d[p] : 0.f;
}

// ---------------------------------------------------------------------------
// host side
// ---------------------------------------------------------------------------
static void launch_gemm(int ta, int tb,
                        const float* A, long sA, int lda,
                        const float* B, long sB, int ldb,
                        float* Cp, long sC, int ldc,
                        int M, int N, int K, int batch, hipStream_t st) {
  dim3 g((N + 31) / 32, (M + 31) / 32, batch), b(32);
  if (!ta && !tb)      gemm_wmma_t<false, false><<<g, b, 0, st>>>(A, sA, lda, B, sB, ldb, Cp, sC, ldc, M, N, K);
  else if (!ta && tb)  gemm_wmma_t<false, true ><<<g, b, 0, st>>>(A, sA, lda, B, sB, ldb, Cp, sC, ldc, M, N, K);
  else                 gemm_wmma_t<true,  false><<<g, b, 0, st>>>(A, sA, lda, B, sB, ldb, Cp, sC, ldc, M, N, K);
}

extern "C" void kernel_launch(void* const* d_in, const int* in_sizes, int n_in,
                              void* d_out, int out_size, void* d_ws, size_t ws_size,
                              hipStream_t stream) {
  const float* X   = (const float*)d_in[0];   // (32, 750, 200) f32
  const float* lab = (const float*)d_in[1];   // (200, 1) f32
  float* out = (float*)d_out;
  float* ws  = (float*)d_ws;

  // ---- scratch layout (floats); total ~49.6M floats (~198 MB) ----
  const long O_XP    = 0;                        // 24,576,000  (reused pool in EM)
  const long O_RCOV  = O_XP    + (long)NM*KC*TP; //  5,120,000  (R, then Cw, then junk)
  const long O_WH    = O_RCOV  + (long)NM*KC2;   //     25,600
  const long O_WHINV = O_WH    + KC2;            //     25,600
  const long O_TMP   = O_WHINV + KC2;            //  5,120,000
  const long O_VB    = O_TMP   + (long)NM*KC2;   //  5,120,000
  const long O_WB    = O_VB    + (long)NM*KC2;   //     32,000
  const long O_RTR   = O_WB    + (long)NM*KC;    //  5,120,000
  const long O_GRAW  = O_RTR   + (long)NM*KC2;   //  4,096,000
  const long O_PSI   = O_GRAW  + (long)KC*KC2;
  const long O_PSIN  = O_PSI   + KC2;
  const long O_U     = O_PSIN  + KC2;
  const long O_UN    = O_U     + KC2;
  const long O_UU    = O_UN    + KC2;
  const long O_PHIS  = O_UU    + KC2;
  const long O_UC    = O_PHIS  + KC2;
  const long O_SIG   = O_UC    + KC2;            // 40,000
  const long O_SYI   = O_SIG   + NM*NM;          // 40,000
  const long O_CHOL  = O_SYI   + NM*NM;          // 40,000
  const long O_Y2    = O_CHOL  + NM*NM;          // 256
  const long O_EIGA  = O_Y2    + 256;            // 25,600
  const long O_EIGV  = O_EIGA  + KC2;            // 25,600
  const long O_EIGW  = O_EIGV  + KC2;            // 256
  const long O_SCAL  = O_EIGW  + 256;            // 16

  // EM big buffers live in the (freed) XP pool
  const long O_RTP  = O_XP;
  const long O_SR   = O_XP + (long)NM*KC2;
  const long O_G    = O_SR + (long)NM*KC2;
  const long O_T1   = O_G  + (long)KC*KC2;
  const long O_PHIB = O_T1 + (long)KC*KC2;
  const long O_QLW  = O_XP;                      // logm stage (after whitening)

  float* XP   = ws + O_XP;   float* RCOV = ws + O_RCOV; float* WH   = ws + O_WH;
  float* WHI  = ws + O_WHINV;float* TMP  = ws + O_TMP;  float* VB   = ws + O_VB;
  float* WB   = ws + O_WB;   float* RTR  = ws + O_RTR;  float* GRAW = ws + O_GRAW;
  float* PSI  = ws + O_PSI;  float* PSIN = ws + O_PSIN; float* U    = ws + O_U;
  float* UN   = ws + O_UN;   float* UU   = ws + O_UU;   float* PHIS = ws + O_PHIS;
  float* UC   = ws + O_UC;   float* SIG  = ws + O_SIG;  float* SYI  = ws + O_SYI;
  float* CHOL = ws + O_CHOL; float* Y2   = ws + O_Y2;   float* EIGA = ws + O_EIGA;
  float* EIGV = ws + O_EIGV; float* EIGW = ws + O_EIGW; float* SCAL = ws + O_SCAL;
  float* RTP = ws + O_RTP; float* SR = ws + O_SR; float* G = ws + O_G;
  float* T1 = ws + O_T1; float* PHIB = ws + O_PHIB; float* QLW = ws + O_QLW;

  const long TOT_XP = (long)NM * KC * TP;
  (void)in_sizes; (void)n_in; (void)out_size; (void)ws_size;

  // ---- enhanced covariance ----
  pack_xhat<<<(unsigned)((TOT_XP + 255) / 256), 256, 0, stream>>>(X, XP);
  // R_m = Xp_m * Xp_m^T   (the dominant GEMM: 200 x [160x768 @ 768x160])
  launch_gemm(0, 1, XP, (long)KC * TP, TP, XP, (long)KC * TP, TP,
              RCOV, KC2, KC, KC, KC, TP, NM, stream);
  trace_scale<<<NM, 256, 0, stream>>>(RCOV);
  mean_wh<<<100, 256, 0, stream>>>(RCOV, WH);

  // Wh^{-1/2} via eigh
  copy_k<<<100, 256, 0, stream>>>(WH, EIGA, KC2);
  jacobi_eigh<<<1, 256, 0, stream>>>(EIGA, EIGV, EIGW, KC, JAC_SWEEPS);
  whinvhalf<<<100, 256, 0, stream>>>(EIGV, EIGW, WHI);

  // whiten: temp_m = Whinv @ R_m @ Whinv ; Cw = sym(temp)
  launch_gemm(0, 0, WHI, 0, KC, RCOV, KC2, KC, XP, KC2, KC, KC, KC, KC, NM, stream);
  launch_gemm(0, 0, XP, KC2, KC, WHI, 0, KC, TMP, KC2, KC, KC, KC, KC, NM, stream);
  symavg<<<(unsigned)(((long)NM * KC2 + 255) / 256), 256, 0, stream>>>(TMP, RCOV, NM);

  // batched eigh of Cw, then matrix log: R_train = Q diag(log w) Q^T
  jacobi_eigh<<<NM, 256, 0, stream>>>(RCOV, VB, WB, KC, JAC_SWEEPS);
  qlw_scale<<<(unsigned)(((long)NM * KC2 + 255) / 256), 256, 0, stream>>>(VB, WB, QLW);
  launch_gemm(0, 1, QLW, KC2, KC, VB, KC2, KC, RTR, KC2, KC, KC, KC, KC, NM, stream);

  // Graw_j = R3[:,j,:]^T @ R3[:,j,:]  (iteration-invariant, for theta)
  // batch over j (KC batches), K = NM trials
  launch_gemm(1, 0, RTR, KC, KC2, RTR, KC, KC2, GRAW, KC2, KC, KC, KC, NM, KC, stream);

  // ---- EM ----
  init_em<<<100, 256, 0, stream>>>(PSI, U, SCAL);
  for (int it = 0; it < NIT; ++it) {
    zero_iter<<<1, 1, 0, stream>>>(SCAL);
    // RtP_m = R3_m @ Psi
    launch_gemm(0, 0, RTR, KC2, KC, PSI, 0, KC, RTP, KC2, KC, KC, KC, KC, NM, stream);
    // Sigma = RtP_flat @ R3_flat^T + lam I  (M=N=200, clamped+store-guarded)
    launch_gemm(0, 1, RTP, 0, KC2, RTR, 0, KC2, SIG, 0, NM, NM, NM, KC2, 1, stream);
    add_lam_diag<<<1, 256, 0, stream>>>(SIG, SCAL);
    chol_invert<<<1, 256, 0, stream>>>(SIG, CHOL, SYI, SCAL);
    y2_loss<<<1, 256, 0, stream>>>(SYI, lab, Y2, SCAL);
    uc_k<<<100, 256, 0, stream>>>(RTP, Y2, UC);
    unew_k<<<100, 256, 0, stream>>>(UC, UN);
    // SR = SyInv @ R_train   (M=200)
    launch_gemm(0, 0, SYI, 0, NM, RTR, 0, KC2, SR, 0, KC2, NM, KC2, NM, 1, stream);
    // G_j = R3[:,j,:]^T @ SR[:,j,:]   (batch over j, K = NM)
    launch_gemm(1, 0, RTR, KC, KC2, SR, KC, KC2, G, KC2, KC, KC, KC, NM, KC, stream);
    // Phi_j = Psi - Psi @ G_j @ Psi
    launch_gemm(0, 0, PSI, 0, KC, G, KC2, KC, T1, KC2, KC, KC, KC, KC, KC, stream);
    launch_gemm(0, 0, T1, KC2, KC, PSI, 0, KC, PHIB, KC2, KC, KC, KC, KC, KC, stream);
    phi_fin<<<(unsigned)(((long)KC * KC2 + 255) / 256), 256, 0, stream>>>(PSI, PHIB);
    theta_k<<<(unsigned)(((long)KC * KC2 + 255) / 256), 256, 0, stream>>>(PHIB, GRAW, SCAL);
    phisum_k<<<100, 256, 0, stream>>>(PHIB, PHIS);
    // UU = Unew @ Unew^T ; Psi_new
    launch_gemm(0, 1, UN, 0, KC, UN, 0, KC, UU, 0, KC, KC, KC, KC, 1, stream);
    psinew_k<<<100, 256, 0, stream>>>(UU, PHIS, PSIN);
    resid_k<<<NM, 256, 0, stream>>>(RTR, UN, lab, SCAL);
    finalize_iter<<<1, 1, 0, stream>>>(SCAL);
    commit_mats<<<100, 256, 0, stream>>>(PSI, PSIN, U, UN, SCAL);
  }

  // ---- final eigh of W = U, threshold, outputs ----
  copy_k<<<100, 256, 0, stream>>>(U, EIGA, KC2);
  jacobi_eigh<<<1, 256, 0, stream>>>(EIGA, EIGV, EIGW, KC, JAC_SWEEPS);
  maxabs_k<<<1, 256, 0, stream>>>(EIGW, SCAL);
  final_out<<<100, 256, 0, stream>>>(U, WH, EIGW, EIGV, SCAL, out);
}